// LiLTPairwiseHead_15977278341601
// MI455X (gfx1250) — compile-verified
//
#include <hip/hip_runtime.h>
#include <hip/hip_bf16.h>

// ---------------------------------------------------------------------------
// LiLT pairwise head:
//   proj = hs(2048x768) @ [W_i^T | W_j^T](768x96)   (bias folded into proj_i)
//   out[b,i,j,c] = proj_i[b*512+i, c] + proj_j[b*512+j, c]
// Phase 1: WMMA f32 16x16x4 GEMM (trivial cost, ~0.3 GFLOP)
// Phase 2: streaming broadcast-add, bound by the 201 MB output write.
// ---------------------------------------------------------------------------

typedef __attribute__((ext_vector_type(2))) float v2f;
typedef __attribute__((ext_vector_type(4))) float v4f;
typedef __attribute__((ext_vector_type(8))) float v8f;

// ---------------------------------------------------------------------------
// Phase 1: proj[2048][96] = hs @ B, where B[:, 0:48] = W_i^T, B[:, 48:96] = W_j^T.
// One wave per 16x16 output tile: 128 M-tiles x 6 N-tiles = 768 waves.
// Launch: 96 blocks x 256 threads (8 waves/block) -> exactly 768 waves, EXEC all 1s.
// ---------------------------------------------------------------------------
__global__ void __launch_bounds__(256)
lilt_proj_gemm(const float* __restrict__ hs,   // (4,512,768) flat = (2048,768)
               const float* __restrict__ W,    // (48,1536)
               const float* __restrict__ bias, // (48,)
               float* __restrict__ proj)       // (2048,96)
{
    const int wave = (blockIdx.x * blockDim.x + threadIdx.x) >> 5;
    const int lane = threadIdx.x & 31;
    const int tileM = wave / 6;
    const int tileN = wave - tileM * 6;
    const int mBase = tileM << 4;
    const int nBase = tileN << 4;

    const int hl  = lane >> 4;   // lane half: 0 or 1
    const int l15 = lane & 15;

    // A (16x4 f32): lane holds row m = mBase + l15; VGPR v holds K = 2*hl + v
    const float* __restrict__ arow = hs + (size_t)(mBase + l15) * 768 + 2 * hl;

    // B (4x16 f32): lane holds col n = nBase + l15; VGPR v holds K = 2*hl + v
    // Column n of B: n<48 -> W_i row n (W[n][k]); n>=48 -> W_j row n-48 (W[n-48][768+k])
    const int n = nBase + l15;
    const size_t wbase = (n < 48) ? (size_t)n * 1536
                                  : (size_t)(n - 48) * 1536 + 768;
    const float* __restrict__ bcol = W + wbase + 2 * hl;

    v8f acc = {};
    for (int k = 0; k < 768; k += 4) {
        v2f a, b;
        a.x = arow[k + 0];
        a.y = arow[k + 1];
        b.x = bcol[k + 0];
        b.y = bcol[k + 1];
        acc = __builtin_amdgcn_wmma_f32_16x16x4_f32(
            /*neg_a=*/false, a, /*neg_b=*/false, b,
            /*c_mod=*/(short)0, acc, /*reuse_a=*/false, /*reuse_b=*/false);
    }

    // Fold bias into proj_i columns only (logits add bias once per c).
    const float bn = (n < 48) ? bias[n] : 0.0f;

    // D (16x16 f32): VGPR v, lane -> row m = mBase + v + 8*hl, col n
#pragma unroll
    for (int v = 0; v < 8; ++v) {
        proj[(size_t)(mBase + v + 8 * hl) * 96 + n] = acc[v] + bn;
    }
}

// ---------------------------------------------------------------------------
// Phase 2: out[mi][j][c] = proj_i[mi][c] + proj_j[b*512+j][c]
// Block tile: TI=4 consecutive i-rows x TJ=64 j-values (same batch b).
// proj_j tile (64 rows x 48 f32 = 12 KB) + proj_i rows staged in LDS once.
// 12 iterations of fully-coalesced 4 KB NT stores per block.
// Grid: (2048/4) * (512/64) = 4096 blocks x 256 threads.
// ---------------------------------------------------------------------------
#define TI 4
#define TJ 64

__global__ void __launch_bounds__(256)
lilt_pairwise_add(const float* __restrict__ proj,  // (2048,96): [:,:48]=i+bias, [:,48:]=j
                  float* __restrict__ out)         // (4,512,512,48)
{
    __shared__ v4f sj[TJ * 12];  // proj_j tile, 12 float4 per row
    __shared__ v4f si[TI * 12];  // proj_i rows

    const int mi0   = (blockIdx.x >> 3) * TI;      // first global i-row (b*512+i)
    const int jBase = (blockIdx.x & 7) * TJ;
    const int b     = mi0 >> 9;                    // batch (TI divides 512)
    const int t     = threadIdx.x;

    const v4f* __restrict__ proj4 = (const v4f*)proj;  // 24 float4 per row

    // Stage proj_j rows jBase..jBase+63 of batch b
#pragma unroll
    for (int q = t; q < TJ * 12; q += 256) {
        const int jl = q / 12, c4 = q - jl * 12;
        sj[q] = proj4[(size_t)(b * 512 + jBase + jl) * 24 + 12 + c4];
    }
    // Stage proj_i rows mi0..mi0+3 (bias already folded in)
    if (t < TI * 12) {
        const int il = t / 12, c4 = t - il * 12;
        si[t] = proj4[(size_t)(mi0 + il) * 24 + c4];
    }
    __syncthreads();

    v4f* __restrict__ out4 = (v4f*)out;  // 12 float4 per (b,i,j) cell

#pragma unroll
    for (int r = 0; r < (TI * TJ * 12) / 256; ++r) {   // 12 iterations
        const int q   = r * 256 + t;
        const int il  = q / (TJ * 12);
        const int rem = q - il * (TJ * 12);
        const int jl  = rem / 12;
        const int c4  = rem - jl * 12;

        const v4f o = si[il * 12 + c4] + sj[rem];

        const size_t oidx = ((size_t)(mi0 + il) * 512 + (jBase + jl)) * 12 + c4;
        __builtin_nontemporal_store(o, &out4[oidx]);
    }
}

// ---------------------------------------------------------------------------
extern "C" void kernel_launch(void* const* d_in, const int* in_sizes, int n_in,
                              void* d_out, int out_size, void* d_ws, size_t ws_size,
                              hipStream_t stream) {
    const float* hs   = (const float*)d_in[0];  // (4,512,768)
    const float* W    = (const float*)d_in[1];  // (48,1536)
    const float* bias = (const float*)d_in[2];  // (48,)
    float* proj = (float*)d_ws;                 // (2048,96) = 786,432 B scratch
    float* out  = (float*)d_out;                // (4,512,512,48)

    lilt_proj_gemm<<<96, 256, 0, stream>>>(hs, W, bias, proj);
    lilt_pairwise_add<<<4096, 256, 0, stream>>>(proj, out);
}